// TripletLoss_74801150427887
// MI455X (gfx1250) — compile-verified
//
#include <hip/hip_runtime.h>
#include <hip/hip_bf16.h>

// Triplet loss over all cyclic shifts == full 1024x1024 pairwise L2 distance
// matrix. Core compute: Gram matrix G = A' * B^T (A' = A + eps) via
// V_WMMA_F32_16X16X4_F32 (exact fp32 WMMA path on CDNA5), then epilogue
// S = |a'|^2 + |b|^2 - 2G, loss = mean relu(d_ap - sqrt(S) + margin), k != j.

#define B_SZ   1024
#define D_SZ   128
#define MARGIN 1.0f
#define EPS_C  1e-6f

typedef __attribute__((ext_vector_type(2))) float v2f;
typedef __attribute__((ext_vector_type(8))) float v8f;

__device__ __forceinline__ float wave_reduce_add(float v) {
#pragma unroll
    for (int off = 16; off >= 1; off >>= 1)
        v += __shfl_xor(v, off, 32);
    return v;
}

// ---------------------------------------------------------------------------
// Kernel 1: per-row stats. One wave32 per row (8 rows / 256-thread block).
//   na2[j] = ||a_j + eps||^2,  nb2[j] = ||b_j||^2,  dap[j] = ||a_j - b_j + eps||
// ---------------------------------------------------------------------------
__global__ void tl_rowstats(const float* __restrict__ A,
                            const float* __restrict__ Bf,
                            float* __restrict__ na2,
                            float* __restrict__ nb2,
                            float* __restrict__ dap) {
    const int wave = threadIdx.x >> 5;
    const int lane = threadIdx.x & 31;
    const int row  = blockIdx.x * 8 + wave;
    const float* pa = A  + row * D_SZ;
    const float* pb = Bf + row * D_SZ;

    float sa = 0.f, sb = 0.f, sd = 0.f;
#pragma unroll
    for (int t = 0; t < D_SZ / 32; ++t) {
        const int d = lane + t * 32;
        const float av = pa[d] + EPS_C;   // a' = a + eps
        const float bv = pb[d];
        const float dv = av - bv;         // = a - b + eps
        sa += av * av;
        sb += bv * bv;
        sd += dv * dv;
    }
    sa = wave_reduce_add(sa);
    sb = wave_reduce_add(sb);
    sd = wave_reduce_add(sd);
    if (lane == 0) {
        na2[row] = sa;
        nb2[row] = sb;
        dap[row] = sqrtf(sd);
    }
}

// ---------------------------------------------------------------------------
// Kernel 2: 16x16 Gram tiles via V_WMMA_F32_16X16X4_F32 (K=128 -> 32 WMMAs),
// fused distance + margin epilogue, deterministic block partial sums.
//
// ISA fp32 operand layouts (CDNA5 ISA 7.12.2):
//   A (16x4): lane l -> row m = l&15, K = 2*(l>>4) + comp   (float2 load)
//   B (4x16): lane l -> col n = l&15, K = 2*(l>>4) + comp   (mirror of A)
//   C/D (16x16): vgpr r, lane l -> m = r + 8*(l>>4), n = l&15
// B fragment loaded from feature_b rows == B^T operand, exactly what the
// Gram matrix needs.
// ---------------------------------------------------------------------------
__global__ void tl_tiles(const float* __restrict__ A,
                         const float* __restrict__ Bf,
                         const float* __restrict__ na2,
                         const float* __restrict__ nb2,
                         const float* __restrict__ dap,
                         float* __restrict__ partials) {
    __shared__ float red[8];

    const int wave = threadIdx.x >> 5;
    const int lane = threadIdx.x & 31;
    const int gid  = blockIdx.x * 8 + wave;      // 0 .. 4095 tiles
    const int ti   = gid >> 6;                   // 64 tiles per dim
    const int tj   = gid & 63;

    const int half = lane >> 4;                  // 0 | 1
    const int l16  = lane & 15;

    const float* pa = A  + (ti * 16 + l16) * D_SZ + half * 2;
    const float* pb = Bf + (tj * 16 + l16) * D_SZ + half * 2;

    v8f acc = {};
#pragma unroll
    for (int kk = 0; kk < D_SZ / 4; ++kk) {      // 32 x v_wmma_f32_16x16x4_f32
        const float2 a2 = *reinterpret_cast<const float2*>(pa + kk * 4);
        const float2 b2 = *reinterpret_cast<const float2*>(pb + kk * 4);
        v2f va; va.x = a2.x + EPS_C; va.y = a2.y + EPS_C;
        v2f vb; vb.x = b2.x;         vb.y = b2.y;
        acc = __builtin_amdgcn_wmma_f32_16x16x4_f32(
            /*neg_a=*/false, va, /*neg_b=*/false, vb,
            /*c_mod=*/(short)0, acc, /*reuse_a=*/false, /*reuse_b=*/false);
    }

    const int   ng = tj * 16 + l16;
    const float nB = nb2[ng];

    float sum = 0.f;
#pragma unroll
    for (int r = 0; r < 8; ++r) {
        const int   mg = ti * 16 + half * 8 + r;
        float S = na2[mg] + nB - 2.f * acc[r];
        S = fmaxf(S, 0.f);
        const float dan = sqrtf(S);
        const float t   = dap[mg] - dan + MARGIN;
        if (mg != ng && t > 0.f) sum += t;       // diagonal (k==j) excluded
    }

    sum = wave_reduce_add(sum);
    if (lane == 0) red[wave] = sum;
    __syncthreads();
    if (threadIdx.x == 0) {
        float s = 0.f;
#pragma unroll
        for (int w = 0; w < 8; ++w) s += red[w];
        partials[blockIdx.x] = s;
    }
}

// ---------------------------------------------------------------------------
// Kernel 3: reduce 512 block partials -> scalar mean. Single 256-thread block,
// fixed-order tree => deterministic.
// ---------------------------------------------------------------------------
__global__ void tl_final(const float* __restrict__ partials,
                         float* __restrict__ out) {
    __shared__ float red[256];
    const int t = threadIdx.x;
    red[t] = partials[t] + partials[t + 256];
    __syncthreads();
#pragma unroll
    for (int off = 128; off >= 1; off >>= 1) {
        if (t < off) red[t] += red[t + off];
        __syncthreads();
    }
    if (t == 0)
        out[0] = red[0] / ((float)B_SZ * (float)(B_SZ - 1));
}

// ---------------------------------------------------------------------------
extern "C" void kernel_launch(void* const* d_in, const int* in_sizes, int n_in,
                              void* d_out, int out_size, void* d_ws, size_t ws_size,
                              hipStream_t stream) {
    const float* a = (const float*)d_in[0];   // [1024,128] fp32
    const float* b = (const float*)d_in[1];   // [1024,128] fp32
    float* out = (float*)d_out;               // scalar loss

    float* ws       = (float*)d_ws;
    float* na2      = ws;                     // 1024
    float* nb2      = ws + B_SZ;              // 1024
    float* dap      = ws + 2 * B_SZ;          // 1024
    float* partials = ws + 3 * B_SZ;          // 512

    tl_rowstats<<<B_SZ / 8, 256, 0, stream>>>(a, b, na2, nb2, dap);
    tl_tiles<<<512, 256, 0, stream>>>(a, b, na2, nb2, dap, partials);
    tl_final<<<1, 256, 0, stream>>>(partials, out);
}